// ElementBlock2D_Lin_69896297775290
// MI455X (gfx1250) — compile-verified
//
#include <hip/hip_runtime.h>
#include <stdint.h>

#define TILE 256

// ---------------------------------------------------------------------------
// CDNA5 async global->LDS copies (ASYNCcnt path, ISA §10.7 / CDNA5 ISA).
// Inline asm is used (portable across ROCm7.2 / amdgpu-toolchain builtin
// arity differences). VDST operand = 32-bit LDS byte offset (low 32 bits of
// the flat shared-aperture address), VADDR = 64-bit global address.
// ---------------------------------------------------------------------------
__device__ __forceinline__ void async_copy_b64(void* lds, const void* g) {
  uint32_t l = (uint32_t)(uintptr_t)lds;
  asm volatile("global_load_async_to_lds_b64 %0, %1, off"
               :: "v"(l), "v"((uint64_t)(uintptr_t)g) : "memory");
}
__device__ __forceinline__ void async_copy_b32(void* lds, const void* g) {
  uint32_t l = (uint32_t)(uintptr_t)lds;
  asm volatile("global_load_async_to_lds_b32 %0, %1, off"
               :: "v"(l), "v"((uint64_t)(uintptr_t)g) : "memory");
}

// ---------------------------------------------------------------------------
// Pass 1: per-element 3x3 shape-function matrix table.
// Row = [m00 m01 m02 m10 m11 m12 m20 m21 m22 pad pad pad] (48B, 16B-aligned).
// Reference's exact division ordering (500K elements -> divides are free);
// also removes the 8x average per-point recomputation.
// ---------------------------------------------------------------------------
__global__ void __launch_bounds__(256) build_M_kernel(
    const int* __restrict__ conn, const float2* __restrict__ coords,
    float* __restrict__ M, int n_elems) {
  int e = blockIdx.x * blockDim.x + threadIdx.x;
  if (e >= n_elems) return;
  int b = e * 3;
  int n1 = conn[b + 0] - 1;
  int n2 = conn[b + 1] - 1;
  int n3 = conn[b + 2] - 1;
  float2 p1 = coords[n1];
  float2 p2 = coords[n2];
  float2 p3 = coords[n3];
  float x1 = p1.x, y1 = p1.y, x2 = p2.x, y2 = p2.y, x3 = p3.x, y3 = p3.y;
  float det = x1 * (y3 - y2) + x2 * (y1 - y3) + x3 * (y2 - y1);
  float nd = -det;
  float4 q0, q1;
  q0.x = (y3 - y2) / det;             // m00
  q0.y = (y1 - y3) / det;             // m01
  q0.z = (y1 - y2) / nd;              // m02
  q0.w = (x2 - x3) / det;             // m10
  q1.x = (x1 - x3) / nd;              // m11
  q1.y = (x1 - x2) / det;             // m12
  q1.z = (x3 * y2 - x2 * y3) / det;   // m20
  q1.w = (x3 * y1 - x1 * y3) / nd;    // m21
  float m22 = (x2 * y1 - x1 * y2) / det;
  float* r = M + (size_t)e * 12;
  ((float4*)r)[0] = q0;
  ((float4*)r)[1] = q1;
  r[8] = m22;
}

// ---------------------------------------------------------------------------
// Pass 2: async double-buffered pipeline.
//  - Next tile's x/cell_id stream into LDS via GLOBAL_LOAD_ASYNC_TO_LDS
//    (ASYNCcnt) while this tile's random, L2-resident M-row gathers
//    (LOADcnt) are in flight.
//  - Exactly 2 async issues per wave per tile (indices clamped, never
//    branched) -> fixed "s_wait_asynccnt 2" leaves only the next tile's
//    pair outstanding.
//  - Each lane reads only its own async-written slot: per-wave counter
//    ordering, no barriers.
//  - Outputs staged per-wave in LDS (stride-3 writes, coprime to 64 banks)
//    then emitted as three dense 128B/wave stores.
// ---------------------------------------------------------------------------
__device__ __forceinline__ void issue_tile(const float2* __restrict__ x,
                                           const int* __restrict__ cell,
                                           int n_pts, int tile, int tid,
                                           float2* sx, int* sc) {
  int idx = tile * TILE + tid;
  int m = n_pts - 1;
  idx = idx < m ? idx : m;  // clamp: uniform issue count, valid addresses
  async_copy_b64(&sx[tid], x + idx);
  async_copy_b32(&sc[tid], cell + idx);
}

__global__ void __launch_bounds__(256) interp_kernel(
    const float2* __restrict__ x, const int* __restrict__ cell,
    const float* __restrict__ M, float* __restrict__ out,
    int n_pts, int n_tiles) {
  __shared__ float2 s_x[2][TILE];
  __shared__ int s_c[2][TILE];
  __shared__ float s_out[TILE * 3];
  const int tid = threadIdx.x;
  const int lane = tid & 31;
  const int wid = tid >> 5;
  const int out_lim = n_pts * 3;

  int cb = 0;
  issue_tile(x, cell, n_pts, blockIdx.x, tid, s_x[0], s_c[0]);  // prologue

  for (int t = blockIdx.x; t < n_tiles; t += gridDim.x) {
    // Kick off next tile into the other buffer (clamped if past the end).
    issue_tile(x, cell, n_pts, t + gridDim.x, tid, s_x[cb ^ 1], s_c[cb ^ 1]);
    // Wait until only the 2 just-issued copies remain -> current tile ready.
    asm volatile("s_wait_asynccnt 0x2" ::: "memory");

    float2 p = s_x[cb][tid];  // ds_load_b64
    int c = s_c[cb][tid];     // ds_load_b32

    const float4* row = (const float4*)(M + (size_t)c * 12);
    float4 q0 = row[0];                 // global_load_b128 (L2-resident)
    float4 q1 = row[1];
    float m22 = M[(size_t)c * 12 + 8];
    float o0 = fmaf(p.x, q0.x, fmaf(p.y, q0.w, q1.z));  // x*m00+y*m10+m20
    float o1 = fmaf(p.x, q0.y, fmaf(p.y, q1.x, q1.w));  // x*m01+y*m11+m21
    float o2 = fmaf(p.x, q0.z, fmaf(p.y, q1.y, m22));   // x*m02+y*m12+m22

    const int sb = wid * 96;
    s_out[sb + lane * 3 + 0] = o0;
    s_out[sb + lane * 3 + 1] = o1;
    s_out[sb + lane * 3 + 2] = o2;
    const int ob = t * (TILE * 3) + sb;
#pragma unroll
    for (int j = 0; j < 3; ++j) {
      int k = ob + j * 32 + lane;
      if (k < out_lim) out[k] = s_out[sb + j * 32 + lane];
    }
    cb ^= 1;
  }
}

// ---------------------------------------------------------------------------
// Fallback: single-pass full gather chain (used only if workspace < 24 MB).
// ---------------------------------------------------------------------------
__global__ void __launch_bounds__(256) direct_kernel(
    const float2* __restrict__ x, const int* __restrict__ cell,
    const int* __restrict__ conn, const float2* __restrict__ coords,
    float* __restrict__ out, int n_pts, int n_tiles) {
  __shared__ float s_out[TILE * 3];
  const int lane = threadIdx.x & 31;
  const int wid = threadIdx.x >> 5;
  const int out_lim = n_pts * 3;
  for (int t = blockIdx.x; t < n_tiles; t += gridDim.x) {
    int i = t * TILE + threadIdx.x;
    int ip = i + gridDim.x * TILE;
    if (ip < n_pts) {
      __builtin_prefetch(&x[ip], 0, 0);
      __builtin_prefetch(&cell[ip], 0, 0);
    }
    float o0 = 0.f, o1 = 0.f, o2 = 0.f;
    if (i < n_pts) {
      float2 p = x[i];
      int c = cell[i];
      int b = c * 3;
      int n1 = conn[b + 0] - 1;
      int n2 = conn[b + 1] - 1;
      int n3 = conn[b + 2] - 1;
      float2 p1 = coords[n1];
      float2 p2 = coords[n2];
      float2 p3 = coords[n3];
      float x1 = p1.x, y1 = p1.y, x2 = p2.x, y2 = p2.y, x3 = p3.x, y3 = p3.y;
      float det = x1 * (y3 - y2) + x2 * (y1 - y3) + x3 * (y2 - y1);
      float inv = 1.0f / det;
      o0 = fmaf(p.x, (y3 - y2), fmaf(p.y, (x2 - x3), (x3 * y2 - x2 * y3))) * inv;
      o1 = fmaf(p.x, (y1 - y3), fmaf(p.y, (x3 - x1), (x1 * y3 - x3 * y1))) * inv;
      o2 = fmaf(p.x, (y2 - y1), fmaf(p.y, (x1 - x2), (x2 * y1 - x1 * y2))) * inv;
    }
    const int sb = wid * 96;
    s_out[sb + lane * 3 + 0] = o0;
    s_out[sb + lane * 3 + 1] = o1;
    s_out[sb + lane * 3 + 2] = o2;
    const int ob = t * (TILE * 3) + sb;
#pragma unroll
    for (int j = 0; j < 3; ++j) {
      int k = ob + j * 32 + lane;
      if (k < out_lim) out[k] = s_out[sb + j * 32 + lane];
    }
  }
}

extern "C" void kernel_launch(void* const* d_in, const int* in_sizes, int n_in,
                              void* d_out, int out_size, void* d_ws, size_t ws_size,
                              hipStream_t stream) {
  const float2* x      = (const float2*)d_in[0];  // (N_POINTS, 2) f32
  const int*    cell   = (const int*)d_in[1];     // (N_POINTS,)   i32
  const int*    conn   = (const int*)d_in[2];     // (N_ELEMS, 3)  i32
  const float2* coords = (const float2*)d_in[3];  // (N_NODES, 2)  f32
  float* out = (float*)d_out;

  const int n_pts   = in_sizes[1];
  const int n_elems = in_sizes[2] / 3;
  const int n_tiles = (n_pts + TILE - 1) / TILE;
  int grid2 = n_tiles < 4096 ? n_tiles : 4096;  // ~32K wave32s in flight

  const size_t ws_needed = (size_t)n_elems * 12 * sizeof(float);
  if (ws_size >= ws_needed) {
    float* M = (float*)d_ws;
    int grid1 = (n_elems + 255) / 256;
    build_M_kernel<<<grid1, 256, 0, stream>>>(conn, coords, M, n_elems);
    interp_kernel<<<grid2, TILE, 0, stream>>>(x, cell, M, out, n_pts, n_tiles);
  } else {
    direct_kernel<<<grid2, TILE, 0, stream>>>(x, cell, conn, coords, out,
                                              n_pts, n_tiles);
  }
}